// FD_9904194585049
// MI455X (gfx1250) — compile-verified
//
#include <hip/hip_runtime.h>
#include <math.h>

#define IC    64
#define GROUP 8
#define KS    3
#define KS2   9
#define OC    72      // GROUP * KS2
#define NB    8       // batch
#define H_    192
#define W_    192
#define HW    (H_ * W_)
#define BN_EPS 1e-5f

typedef __attribute__((ext_vector_type(2))) float v2f;
typedef __attribute__((ext_vector_type(8))) float v8f;

// ---------------------------------------------------------------------------
// Pass 1: global average pool. One block per (n,c) plane (512 planes).
// Regular-temporal loads on purpose: this primes L2 with x (75.5 MB < 192 MB)
// so pass 3's re-read of x hits L2 instead of HBM.
// ---------------------------------------------------------------------------
__global__ void __launch_bounds__(256)
fd_pool_kernel(const float* __restrict__ x, float* __restrict__ pooled) {
    const int plane = blockIdx.x;                       // n*64 + c
    const float4* px = (const float4*)(x + (size_t)plane * HW);
    float s = 0.0f;
    for (int i = threadIdx.x; i < HW / 4; i += 256) {
        float4 v = px[i];
        s += (v.x + v.y) + (v.z + v.w);
    }
    __shared__ float red[256];
    red[threadIdx.x] = s;
    __syncthreads();
    for (int off = 128; off > 0; off >>= 1) {
        if (threadIdx.x < off) red[threadIdx.x] += red[threadIdx.x + off];
        __syncthreads();
    }
    if (threadIdx.x == 0)
        __builtin_nontemporal_store(red[0] * (1.0f / (float)HW), &pooled[plane]);
}

// ---------------------------------------------------------------------------
// Pass 2: kernel generation. lf_raw = pooled(8x64) @ conv_w^T(64x72), BN,
// softmax over the 9 taps of each (sample, group).
// Single wave32, V_WMMA_F32_16X16X4_F32: M=8 (pad 16), K=64 (16 steps), N=72
// (5 tiles of 16).
// A 16x4 f32 layout: lanes 0-15 -> M=lane, vgpr0/1 = K+0/K+1;
//                    lanes 16-31 -> M=lane-16, vgpr0/1 = K+2/K+3.
// B 4x16 layout mirrors A with N striped across lanes.
// D: vgpr v, lanes 0-15 -> row M=v, col N=lane (rows 0..7 are the valid ones).
// Zero-padding done branch-free (clamped address + mask multiply) so EXEC
// stays all-ones with no saveexec churn around the WMMA feed loads.
// ---------------------------------------------------------------------------
__global__ void __launch_bounds__(32)
fd_kgen_kernel(const float* __restrict__ pooled,   // (8, 64)
               const float* __restrict__ conv_w,   // (72, 64) row-major
               const float* __restrict__ gamma,
               const float* __restrict__ beta,
               const float* __restrict__ mean,
               const float* __restrict__ var,
               float* __restrict__ lf)              // (8, 8, 9) softmaxed
{
    __shared__ float sh[NB * OC];                    // BN'd logits, 8 x 72
    const int lane = threadIdx.x;
    const int half = lane >> 4;                      // 0: K+0/1, 1: K+2/3
    const int l16  = lane & 15;

    const int   arow = (l16 < NB) ? l16 : 0;         // clamped A row
    const float amsk = (l16 < NB) ? 1.0f : 0.0f;     // zero-pad rows 8..15
    const float* arow_p = pooled + arow * IC + half * 2;

    for (int nt = 0; nt < 5; ++nt) {
        const int nbase = nt * 16;
        const int ncol  = nbase + l16;
        const int   bcol = (ncol < OC) ? ncol : 0;   // clamped B column
        const float bmsk = (ncol < OC) ? 1.0f : 0.0f;
        const float* bcol_p = conv_w + bcol * IC + half * 2;

        v8f acc = {};                                // C = 0
#pragma unroll
        for (int kk = 0; kk < 16; ++kk) {
            const int kbase = kk * 4;
            v2f a, b;
            a[0] = amsk * arow_p[kbase];
            a[1] = amsk * arow_p[kbase + 1];
            b[0] = bmsk * bcol_p[kbase];             // B[k][ncol] = W[ncol][k]
            b[1] = bmsk * bcol_p[kbase + 1];
            acc = __builtin_amdgcn_wmma_f32_16x16x4_f32(
                /*neg_a=*/false, a, /*neg_b=*/false, b,
                /*c_mod=*/(short)0, acc, /*reuse_a=*/false, /*reuse_b=*/false);
        }
        // Writeback valid rows (M = vgpr index, lanes 0-15) with fused BN.
        if (half == 0 && ncol < OC) {
            const float sc = rsqrtf(var[ncol] + BN_EPS) * gamma[ncol];
            const float sb = beta[ncol] - mean[ncol] * sc;
#pragma unroll
            for (int v = 0; v < 8; ++v)
                sh[v * OC + ncol] = acc[v] * sc + sb;
        }
    }
    __syncthreads();

    // Softmax over 9 taps for each of 64 (sample, group) pairs.
    for (int p = lane; p < NB * GROUP; p += 32) {
        const int m = p / GROUP, g = p % GROUP;
        const float* row = &sh[m * OC + g * KS2];
        float mx = row[0];
#pragma unroll
        for (int k = 1; k < KS2; ++k) mx = fmaxf(mx, row[k]);
        float e[KS2];
        float sum = 0.0f;
#pragma unroll
        for (int k = 0; k < KS2; ++k) { e[k] = __expf(row[k] - mx); sum += e[k]; }
        const float inv = 1.0f / sum;
#pragma unroll
        for (int k = 0; k < KS2; ++k)
            __builtin_nontemporal_store(e[k] * inv, &lf[p * KS2 + k]);
    }
}

// ---------------------------------------------------------------------------
// Pass 3: reflection-padded 3x3 dynamic weighted sum + residual.
// Block = one image row (192 threads, 1 px/thread, coalesced 4B/lane).
// blockIdx.x = h, blockIdx.y = plane (n*64+c). Weights uniform per block.
// x reads: regular-temporal (expected L2-resident from pass 1).
// low/high writes: NON-temporal — never re-read, must not evict x from L2.
// ---------------------------------------------------------------------------
__global__ void __launch_bounds__(192)
fd_dynfilt_kernel(const float* __restrict__ x,
                  const float* __restrict__ lf,     // (8, 8, 9)
                  float* __restrict__ low,
                  float* __restrict__ high)
{
    const int h     = blockIdx.x;
    const int plane = blockIdx.y;                    // n*64 + c
    const int n     = plane >> 6;
    const int c     = plane & 63;
    const int g     = c >> 3;
    const float* wgt = &lf[((size_t)n * GROUP + g) * KS2];
    const float w0 = wgt[0], w1 = wgt[1], w2 = wgt[2];
    const float w3 = wgt[3], w4 = wgt[4], w5 = wgt[5];
    const float w6 = wgt[6], w7 = wgt[7], w8 = wgt[8];

    const int w  = threadIdx.x;
    const int hm = (h == 0)      ? 1      : h - 1;   // reflect (no edge repeat)
    const int hp = (h == H_ - 1) ? H_ - 2 : h + 1;
    const int wm = (w == 0)      ? 1      : w - 1;
    const int wp = (w == W_ - 1) ? W_ - 2 : w + 1;

    const float* xp = x + (size_t)plane * HW;
    const float* r0 = xp + (size_t)hm * W_;
    const float* r1 = xp + (size_t)h  * W_;
    const float* r2 = xp + (size_t)hp * W_;

    const float ctr = r1[w];
    float acc = w0 * r0[wm] + w1 * r0[w] + w2 * r0[wp]
              + w3 * r1[wm] + w4 * ctr   + w5 * r1[wp]
              + w6 * r2[wm] + w7 * r2[w] + w8 * r2[wp];

    const size_t idx = (size_t)plane * HW + (size_t)h * W_ + w;
    __builtin_nontemporal_store(acc,       &low[idx]);
    __builtin_nontemporal_store(ctr - acc, &high[idx]);
}

// ---------------------------------------------------------------------------
extern "C" void kernel_launch(void* const* d_in, const int* in_sizes, int n_in,
                              void* d_out, int out_size, void* d_ws, size_t ws_size,
                              hipStream_t stream) {
    const float* x      = (const float*)d_in[0];   // (8, 64, 192, 192)
    const float* conv_w = (const float*)d_in[1];   // (72, 64)
    const float* gamma  = (const float*)d_in[2];   // (72,)
    const float* beta   = (const float*)d_in[3];   // (72,)
    const float* mean   = (const float*)d_in[4];   // (72,)
    const float* var    = (const float*)d_in[5];   // (72,)

    float* pooled = (float*)d_ws;                  // 512 floats
    float* lf     = pooled + NB * IC;              // 576 floats

    float* low  = (float*)d_out;
    float* high = low + (size_t)NB * IC * HW;

    fd_pool_kernel<<<NB * IC, 256, 0, stream>>>(x, pooled);
    fd_kgen_kernel<<<1, 32, 0, stream>>>(pooled, conv_w, gamma, beta, mean, var, lf);
    dim3 grid(H_, NB * IC);
    fd_dynfilt_kernel<<<grid, 192, 0, stream>>>(x, lf, low, high);
}